// LinearCRF_77060303224897
// MI455X (gfx1250) — compile-verified
//
#include <hip/hip_runtime.h>
#include <hip/hip_bf16.h>
#include <stdint.h>

// ---------------------------------------------------------------------------
// Linear-chain CRF NLL on gfx1250 (MI455X).
//   B=256, T=2048, C=48.
//   logZ: forward scan as exp-trick matmul -> V_WMMA_F32_16X16X4_F32,
//         emissions streamed by Tensor Data Mover (tensor_load_to_lds),
//         double-buffered in LDS, synced with s_wait_tensorcnt.
//   path score: parallel gather + block reduction (deterministic).
// ---------------------------------------------------------------------------

typedef __attribute__((ext_vector_type(2))) float v2f;
typedef __attribute__((ext_vector_type(8))) float v8f;
typedef unsigned int v4u __attribute__((ext_vector_type(4)));
typedef int v4i __attribute__((ext_vector_type(4)));
typedef int v8i __attribute__((ext_vector_type(8)));

static constexpr int Bsz  = 256;
static constexpr int Tlen = 2048;
static constexpr int C    = 48;
static constexpr int NT   = 3;   // 48 / 16 column tiles
static constexpr int NK   = 12;  // 48 / 4  K chunks
static constexpr int GROUPS = 2; // interleaved 16-row batch groups per wave (ILP)
static constexpr int ROWS = 16 * GROUPS; // 32 batch rows per wave

#define WMMA4(Aop, Bop, Cop) \
    __builtin_amdgcn_wmma_f32_16x16x4_f32(false, (Aop), false, (Bop), (short)0, (Cop), false, false)

#if __has_include(<hip/amd_detail/amd_gfx1250_TDM.h>)
#define TDM_SIX_ARGS 1
#endif

// LDS byte offset of a __shared__ object (generic -> AS(3) addrspacecast)
template <typename T>
__device__ inline uint32_t lds_addr_of(T* p) {
    return (uint32_t)(uintptr_t)(__attribute__((address_space(3))) T*)p;
}

// Issue one TDM 2-D tile load: ROWS x 48 f32, row stride Tlen*C elements,
// from gptr into LDS at lds_byte_off.  D# per CDNA5 ISA ch.8.
__device__ inline void tdm_load_tile(const float* gptr, uint32_t lds_byte_off)
{
    const uint64_t ga = (uint64_t)(uintptr_t)gptr;
    v4u g0;
    g0.x = 1u;                                   // count=1 (valid user descriptor)
    g0.y = lds_byte_off;                         // lds_addr
    g0.z = (uint32_t)(ga & 0xFFFFFFFFu);         // global_addr[31:0]
    g0.w = (uint32_t)((ga >> 32) & 0x01FFFFFFu)  // global_addr[56:32]
         | (2u << 30);                           // type = 2 ("image")
    v8i g1;
    g1[0] = (int)(2u << 16);                     // workgroup_mask=0, data_size=2 (4B)
    g1[1] = (int)((uint32_t)C << 16);            // tensor_dim0 = 48 (lo16)
    g1[2] = (int)((uint32_t)ROWS << 16);         // tensor_dim0 hi16=0 | tensor_dim1 lo16 = 32
    g1[3] = (int)((uint32_t)C << 16);            // tensor_dim1 hi16=0 | tile_dim0 = 48
    g1[4] = (int)ROWS;                           // tile_dim1 = 32 | tile_dim2 = 0
    g1[5] = (int)(Tlen * C);                     // tensor_dim0_stride lo32 = 98304
    g1[6] = 0;                                   // stride hi16 = 0 | tensor_dim1_stride = 0
    g1[7] = 0;
    const v4i z4 = {0, 0, 0, 0};
#ifdef TDM_SIX_ARGS
    const v8i z8 = {0, 0, 0, 0, 0, 0, 0, 0};
    __builtin_amdgcn_tensor_load_to_lds(g0, g1, z4, z4, z8, 0);
#else
    __builtin_amdgcn_tensor_load_to_lds(g0, g1, z4, z4, 0);
#endif
}

// ---------------------------------------------------------------------------
// Forward scan: one wave per block, ROWS batch rows per wave.
// alpha kept in the WMMA C/D register layout:
//   value (row = r + 8*(lane>=16), col = 16*nt + lane%16) lives in alpha[g][nt][r]
// ---------------------------------------------------------------------------
__global__ __launch_bounds__(32, 1)
void crf_forward_scan(const float* __restrict__ emis,
                      const uint8_t* __restrict__ mask,
                      const float* __restrict__ trans,
                      const float* __restrict__ startT,
                      const float* __restrict__ endT,
                      float* __restrict__ logZ)
{
    __shared__ float sExpT[C * C];           // exp(transitions)
    __shared__ float sE[GROUPS][16][C];      // exp(alpha - rowmax), [row][class]
    __shared__ float sEm[2][ROWS][C];        // TDM double buffer: emissions tile for step t

    const int lane = threadIdx.x;            // 0..31 (wave32)
    const int half = lane >> 4;              // 0 / 1
    const int ln   = lane & 15;              // 0..15
    const int b0   = blockIdx.x * ROWS;

    const uint32_t emBase0 = lds_addr_of(&sEm[0][0][0]);
    const uint32_t emBase1 = lds_addr_of(&sEm[1][0][0]);

    // exp(transitions) -> LDS (2304 f32)
    for (int i = lane; i < C * C; i += 32) sExpT[i] = __expf(trans[i]);

    // kick off the TDM pipeline: emissions tile for t=1 into buffer 1
    tdm_load_tile(&emis[((size_t)b0 * Tlen + 1) * C], emBase1);

    __syncthreads();

    // B operands (constant over time), WMMA B 4x16 f32 layout:
    //   VGPR r of chunk kc holds K = 4*kc + r + 2*half, N = ln
    v2f bop[NT][NK];
#pragma unroll
    for (int nt = 0; nt < NT; ++nt) {
#pragma unroll
        for (int kc = 0; kc < NK; ++kc) {
            const int k0  = 4 * kc + 2 * half;
            const int col = 16 * nt + ln;
            bop[nt][kc].x = sExpT[(k0 + 0) * C + col];
            bop[nt][kc].y = sExpT[(k0 + 1) * C + col];
        }
    }

    // end-transition values for this lane's columns (used after the scan)
    float eT[NT];
#pragma unroll
    for (int nt = 0; nt < NT; ++nt) eT[nt] = endT[16 * nt + ln];

    // alpha0 = start + emissions[:, 0]
    float alpha[GROUPS][NT][8];
#pragma unroll
    for (int g = 0; g < GROUPS; ++g) {
#pragma unroll
        for (int nt = 0; nt < NT; ++nt) {
            const int col = 16 * nt + ln;
            const float st = startT[col];
#pragma unroll
            for (int r = 0; r < 8; ++r) {
                const size_t row = (size_t)(b0 + 16 * g + 8 * half + r);
                alpha[g][nt][r] = st + emis[row * (size_t)(Tlen * C) + col];
            }
        }
    }

#pragma unroll 1
    for (int t = 1; t < Tlen; ++t) {
        // one mask byte per lane covers all 32 rows of this wave; ballot bit l == block-row l
        const uint8_t mb = mask[(size_t)(b0 + lane) * Tlen + t];
        const unsigned long long bal = __ballot(mb != 0);

        // async-prefetch emissions tile for t+1, then wait until tile t is resident
        if (t + 1 < Tlen) {
            tdm_load_tile(&emis[((size_t)b0 * Tlen + (t + 1)) * C],
                          ((t + 1) & 1) ? emBase1 : emBase0);
            __builtin_amdgcn_s_wait_tensorcnt(1);   // in-order: tile t complete
        } else {
            __builtin_amdgcn_s_wait_tensorcnt(0);
        }

        float mxs[GROUPS][8];
#pragma unroll
        for (int g = 0; g < GROUPS; ++g) {
            // per-row max over all 48 columns: tile-combine then shfl within 16-lane halves
#pragma unroll
            for (int r = 0; r < 8; ++r) {
                float m = fmaxf(fmaxf(alpha[g][0][r], alpha[g][1][r]), alpha[g][2][r]);
#pragma unroll
                for (int off = 1; off < 16; off <<= 1)
                    m = fmaxf(m, __shfl_xor(m, off, 32));
                mxs[g][r] = m;
            }
            // exp(alpha - rowmax) -> LDS in [row][class] form (transpose staging for A)
#pragma unroll
            for (int nt = 0; nt < NT; ++nt) {
                const int col = 16 * nt + ln;
#pragma unroll
                for (int r = 0; r < 8; ++r)
                    sE[g][8 * half + r][col] = __expf(alpha[g][nt][r] - mxs[g][r]);
            }
        }
        __syncthreads();

        const int buf = t & 1;
#pragma unroll
        for (int g = 0; g < GROUPS; ++g) {
            // A operands, WMMA A 16x4 f32 layout: M = ln, VGPR0 K = 4*kc + 2*half, VGPR1 K+1
            v2f aop[NK];
#pragma unroll
            for (int kc = 0; kc < NK; ++kc) {
                const float* p = &sE[g][ln][4 * kc + 2 * half];
                aop[kc].x = p[0];
                aop[kc].y = p[1];
            }

            v8f acc0 = {0.f, 0.f, 0.f, 0.f, 0.f, 0.f, 0.f, 0.f};
            v8f acc1 = {0.f, 0.f, 0.f, 0.f, 0.f, 0.f, 0.f, 0.f};
            v8f acc2 = {0.f, 0.f, 0.f, 0.f, 0.f, 0.f, 0.f, 0.f};
#pragma unroll
            for (int kc = 0; kc < NK; ++kc) {   // 3-way interleave: no WMMA->WMMA RAW
                acc0 = WMMA4(aop[kc], bop[0][kc], acc0);
                acc1 = WMMA4(aop[kc], bop[1][kc], acc1);
                acc2 = WMMA4(aop[kc], bop[2][kc], acc2);
            }

            // alpha' = em + rowmax + log(acc), gated by mask; em from the TDM-fed LDS tile
#pragma unroll
            for (int nt = 0; nt < NT; ++nt) {
                const v8f acc = (nt == 0) ? acc0 : (nt == 1) ? acc1 : acc2;
                const int col = 16 * nt + ln;
#pragma unroll
                for (int r = 0; r < 8; ++r) {
                    const int rowInBlk = 16 * g + 8 * half + r;
                    const float em  = sEm[buf][rowInBlk][col];
                    const float nv  = em + mxs[g][r] + __logf(acc[r]);
                    const bool live = (bal >> rowInBlk) & 1ull;
                    alpha[g][nt][r] = live ? nv : alpha[g][nt][r];
                }
            }
        }
        __syncthreads();   // protect sE / sEm WAR for next iteration
    }

    // logZ = logsumexp(alpha + end_transitions) per row
#pragma unroll
    for (int g = 0; g < GROUPS; ++g) {
        float res[8];
#pragma unroll
        for (int r = 0; r < 8; ++r) {
            float m = -3.4e38f;
#pragma unroll
            for (int nt = 0; nt < NT; ++nt)
                m = fmaxf(m, alpha[g][nt][r] + eT[nt]);
#pragma unroll
            for (int off = 1; off < 16; off <<= 1)
                m = fmaxf(m, __shfl_xor(m, off, 32));
            float s = 0.f;
#pragma unroll
            for (int nt = 0; nt < NT; ++nt)
                s += __expf(alpha[g][nt][r] + eT[nt] - m);
#pragma unroll
            for (int off = 1; off < 16; off <<= 1)
                s += __shfl_xor(s, off, 32);
            res[r] = m + __logf(s);
        }
        if (ln == 0) {
#pragma unroll
            for (int r = 0; r < 8; ++r)
                logZ[b0 + 16 * g + 8 * half + r] = res[r];
        }
    }
}

// ---------------------------------------------------------------------------
// Path score + per-batch NLL (one block per batch element)
// ---------------------------------------------------------------------------
__global__ __launch_bounds__(256, 4)
void crf_path_nll(const float* __restrict__ emis,
                  const long long* __restrict__ tags,
                  const uint8_t* __restrict__ mask,
                  const float* __restrict__ trans,
                  const float* __restrict__ startT,
                  const float* __restrict__ endT,
                  const float* __restrict__ logZ,
                  float* __restrict__ nll)
{
    __shared__ float sS[256];
    __shared__ int   sC[256];
    const int b = blockIdx.x, tid = threadIdx.x;

    float s = 0.f;
    int cnt = 0;
    for (int t = tid; t < Tlen; t += 256) {
        const int m = mask[(size_t)b * Tlen + t] ? 1 : 0;
        cnt += m;
        const long long tg = tags[(size_t)b * Tlen + t];
        if (t == 0) {
            s += startT[tg] + emis[(size_t)b * (size_t)(Tlen * C) + tg];
        } else if (m) {
            const long long tp = tags[(size_t)b * Tlen + t - 1];
            s += emis[(size_t)b * (size_t)(Tlen * C) + (size_t)t * C + tg]
               + trans[(size_t)tp * C + tg];
        }
    }
    sS[tid] = s; sC[tid] = cnt;
    __syncthreads();
    for (int o = 128; o > 0; o >>= 1) {
        if (tid < o) { sS[tid] += sS[tid + o]; sC[tid] += sC[tid + o]; }
        __syncthreads();
    }
    if (tid == 0) {
        const int last = sC[0] - 1;
        const long long lt = tags[(size_t)b * Tlen + last];
        const float ps = sS[0] + endT[lt];
        nll[b] = logZ[b] - ps;
    }
}

// Deterministic final mean (no float atomics)
__global__ __launch_bounds__(256, 1)
void crf_mean(const float* __restrict__ nll, float* __restrict__ out)
{
    __shared__ float sS[256];
    const int tid = threadIdx.x;
    sS[tid] = nll[tid];
    __syncthreads();
    for (int o = 128; o > 0; o >>= 1) {
        if (tid < o) sS[tid] += sS[tid + o];
        __syncthreads();
    }
    if (tid == 0) out[0] = sS[0] * (1.0f / (float)Bsz);
}

// ---------------------------------------------------------------------------
extern "C" void kernel_launch(void* const* d_in, const int* in_sizes, int n_in,
                              void* d_out, int out_size, void* d_ws, size_t ws_size,
                              hipStream_t stream)
{
    const float*     emis   = (const float*)d_in[0];      // [B,T,C] f32
    const long long* tags   = (const long long*)d_in[1];  // [B,T]   i64
    const uint8_t*   mask   = (const uint8_t*)d_in[2];    // [B,T]   bool
    const float*     trans  = (const float*)d_in[3];      // [C,C]
    const float*     startT = (const float*)d_in[4];      // [C]
    const float*     endT   = (const float*)d_in[5];      // [C]
    float*           out    = (float*)d_out;              // scalar

    float* logZ = (float*)d_ws;          // [B]
    float* nll  = logZ + Bsz;            // [B]

    crf_forward_scan<<<dim3(Bsz / ROWS), dim3(32), 0, stream>>>(
        emis, mask, trans, startT, endT, logZ);
    crf_path_nll<<<dim3(Bsz), dim3(256), 0, stream>>>(
        emis, tags, mask, trans, startT, endT, logZ, nll);
    crf_mean<<<dim3(1), dim3(256), 0, stream>>>(nll, out);
}